// CRF_9414568312957
// MI455X (gfx1250) — compile-verified
//
#include <hip/hip_runtime.h>
#include <hip/hip_bf16.h>

typedef float v2f __attribute__((ext_vector_type(2)));
typedef float v8f __attribute__((ext_vector_type(8)));

namespace {
constexpr float kImpossible = -10000.0f;
constexpr int kC     = 52;   // NUM_TAGS + 2 (incl. start/stop)
constexpr int kCPad  = 64;   // padded tag dim = 4 WMMA N-tiles
constexpr int kF     = 768;
constexpr int kB     = 128;
constexpr int kL     = 512;
constexpr int kStart = kC - 2;  // 50
constexpr int kStop  = kC - 1;  // 51
constexpr int kSlices = 4;      // 52 = 4 x 13 'from' values per slice
constexpr int kFPer   = kC / kSlices;
}

// ---------------------------------------------------------------------------
// Kernel 1: emit[m][c] = sum_k feat[m][k] * W[c][k] + b[c]    (m = b*L + t)
// One wave (32 threads) per 16-row M tile; computes all 4 N tiles (C pad 64)
// with V_WMMA_F32_16X16X4_F32 so features stream through HBM exactly once
// (201 MB @ 23.3 TB/s ~= 9 us floor; fp32 WMMA keeps argmax-exact precision).
//
// fp32 WMMA operand layout (wave32):
//   A (16x4, MxK):  lanes 0-15: M=lane, {v0,v1}={K+0,K+1}; lanes 16-31: {K+2,K+3}
//   B (4x16, KxN):  lanes 0-15: N=lane, {v0,v1}={K+0,K+1}; lanes 16-31: {K+2,K+3}
//   C/D (16x16):    v[i]: lanes 0-15 -> M=i, N=lane; lanes 16-31 -> M=i+8
// ---------------------------------------------------------------------------
__global__ __launch_bounds__(32)
void crf_emit_wmma(const float* __restrict__ feat,
                   const float* __restrict__ W,
                   const float* __restrict__ bias,
                   float* __restrict__ emit) {
  const int lane = threadIdx.x;
  const int m0   = blockIdx.x * 16;
  const int r    = lane & 15;
  const int ks   = (lane >> 4) * 2;   // 0 for lanes 0-15, 2 for lanes 16-31

  const float* A  = feat + (size_t)(m0 + r) * kF + ks;
  const float* B0 = W + (size_t)(r)      * kF + ks;
  const float* B1 = W + (size_t)(16 + r) * kF + ks;
  const float* B2 = W + (size_t)(32 + r) * kF + ks;
  const bool   v3 = (48 + r) < kC;                       // rows 52..63 are padding
  const float* B3 = W + (size_t)(v3 ? (48 + r) : 0) * kF + ks;

  v8f acc0 = {}, acc1 = {}, acc2 = {}, acc3 = {};

  for (int k = 0; k < kF; k += 4) {
    __builtin_prefetch(A + k + 64, 0, 0);   // global_prefetch_b8, one line ahead
    v2f a  = *(const v2f*)(A  + k);
    v2f b0 = *(const v2f*)(B0 + k);
    v2f b1 = *(const v2f*)(B1 + k);
    v2f b2 = *(const v2f*)(B2 + k);
    v2f b3 = *(const v2f*)(B3 + k);
    if (!v3) { b3.x = 0.0f; b3.y = 0.0f; }  // zero padded W rows

    acc0 = __builtin_amdgcn_wmma_f32_16x16x4_f32(false, a, false, b0, (short)0, acc0, false, false);
    acc1 = __builtin_amdgcn_wmma_f32_16x16x4_f32(false, a, false, b1, (short)0, acc1, false, false);
    acc2 = __builtin_amdgcn_wmma_f32_16x16x4_f32(false, a, false, b2, (short)0, acc2, false, false);
    acc3 = __builtin_amdgcn_wmma_f32_16x16x4_f32(false, a, false, b3, (short)0, acc3, false, false);
  }

  const int col   = lane & 15;
  const int rbase = (lane >> 4) * 8;
  #pragma unroll
  for (int i = 0; i < 8; ++i) {
    const size_t m = (size_t)(m0 + rbase + i) * kCPad;
    emit[m + col]      = acc0[i] + bias[col];
    emit[m + 16 + col] = acc1[i] + bias[16 + col];
    emit[m + 32 + col] = acc2[i] + bias[32 + col];
    const int n3 = 48 + col;
    emit[m + n3]       = acc3[i] + (n3 < kC ? bias[n3] : 0.0f);  // cols >=52 never read
  }
}

// ---------------------------------------------------------------------------
// Kernel 2: Viterbi recursion + backtrace. One block per batch element.
// 256 threads (8 waves): thread (slice, to) with slice = tid>>6 in [0,4),
// to = tid&63 (< 52 active). The 52-long argmax chain (a serial
// cmp+cndmask dependence the compiler cannot reassociate) is split into
// 4 partial chains of 13, combined in ascending-slice order with strict '>'
// so jnp.argmax first-max tie-breaking is preserved.
// Transposed+padded T in LDS (bank-conflict free: bank == to), ping-pong
// score buffers, all 512x52 backpointers in LDS as u8 (CDNA5: 320KB/WGP),
// so the backtrace's 511 dependent loads are LDS-latency, not HBM-latency.
// ---------------------------------------------------------------------------
__global__ __launch_bounds__(256)
void crf_viterbi(const float* __restrict__ emit,
                 const int*   __restrict__ masks,
                 const float* __restrict__ T,
                 float* __restrict__ best_score_out,
                 int*   __restrict__ paths_out) {
  __shared__ float Tt[kC * kCPad];          // Tt[from*64 + to] = T[to][from]
  __shared__ float sc[2][kCPad];            // ping-pong Viterbi scores
  __shared__ float pv[kSlices][kCPad];      // per-slice partial max
  __shared__ int   pa[kSlices][kCPad];      // per-slice partial argmax
  __shared__ unsigned char bp[kL][kC];      // backpointers, 26.6 KB

  const int tid   = threadIdx.x;
  const int slice = tid >> 6;               // 0..3
  const int to    = tid & 63;               // 0..63, active when < 52

  const int b = blockIdx.x;

  for (int idx = tid; idx < kC * kC; idx += 256) {
    const int t2 = idx / kC, f2 = idx % kC;
    Tt[f2 * kCPad + t2] = T[t2 * kC + f2];
  }
  if (tid < kCPad) sc[0][tid] = (tid == kStart) ? 0.0f : kImpossible;
  __syncthreads();

  const float* eb = emit + (size_t)b * kL * kCPad;
  const int*   mb = masks + (size_t)b * kL;

  int cur = 0;
  for (int t = 0; t < kL; ++t) {
    if (to < kC) {
      const int f0 = slice * kFPer;
      float best = sc[cur][f0] + Tt[f0 * kCPad + to];
      int   arg  = f0;
      #pragma unroll
      for (int j = 1; j < kFPer; ++j) {
        const int f = f0 + j;
        const float v = sc[cur][f] + Tt[f * kCPad + to];
        if (v > best) { best = v; arg = f; }   // first-max within slice
      }
      pv[slice][to] = best;
      pa[slice][to] = arg;
    }
    __syncthreads();
    if (tid < kC) {
      float best = pv[0][tid];
      int   arg  = pa[0][tid];
      #pragma unroll
      for (int s2 = 1; s2 < kSlices; ++s2) {
        const float v = pv[s2][tid];
        if (v > best) { best = v; arg = pa[s2][tid]; }  // earlier slice wins ties
      }
      bp[t][tid] = (unsigned char)arg;
      const float m = (float)mb[t];
      const float e = eb[(size_t)t * kCPad + tid];
      sc[cur ^ 1][tid] = (best + e) * m + sc[cur][tid] * (1.0f - m);
    }
    cur ^= 1;
    __syncthreads();
  }

  if (tid == 0) {
    // final = score + T[stop][c];  best_score / best_tag
    float best = sc[cur][0] + Tt[0 * kCPad + kStop];
    int   tag  = 0;
    for (int c = 1; c < kC; ++c) {
      const float v = sc[cur][c] + Tt[c * kCPad + kStop];
      if (v > best) { best = v; tag = c; }
    }
    best_score_out[b] = best;
    int* pb = paths_out + (size_t)b * kL;
    pb[kL - 1] = tag;
    for (int t = kL - 2; t >= 0; --t) {   // backtrace entirely from LDS
      if (mb[t]) tag = (int)bp[t + 1][tag];
      pb[t] = tag;
    }
  }
}

// ---------------------------------------------------------------------------
// Inputs (setup_inputs order): features f32[128,512,768], masks i32[128,512],
// W f32[52,768], b f32[52], transitions f32[52,52].
// Output tuple: best_score f32[128] ++ best_paths i32[128,512].
// Workspace: padded emit matrix f32[65536,64] = 16 MiB.
// ---------------------------------------------------------------------------
extern "C" void kernel_launch(void* const* d_in, const int* in_sizes, int n_in,
                              void* d_out, int out_size, void* d_ws, size_t ws_size,
                              hipStream_t stream) {
  const float* feat  = (const float*)d_in[0];
  const int*   masks = (const int*)  d_in[1];
  const float* W     = (const float*)d_in[2];
  const float* bias  = (const float*)d_in[3];
  const float* T     = (const float*)d_in[4];

  float* emit       = (float*)d_ws;                     // 65536*64 f32
  float* best_score = (float*)d_out;                    // [128]
  int*   paths      = (int*)((float*)d_out + kB);       // [128*512]

  crf_emit_wmma<<<(kB * kL) / 16, 32, 0, stream>>>(feat, W, bias, emit);
  crf_viterbi<<<kB, 256, 0, stream>>>(emit, masks, T, best_score, paths);
}